// RCNNTargetGenerator_53145925320994
// MI455X (gfx1250) — compile-verified
//
#include <hip/hip_runtime.h>
#include <stdint.h>

// ---------------------------------------------------------------------------
// RCNN target generator (elementwise bbox delta encoding) — MI455X / gfx1250.
//
// Roofline: ~30 FLOP vs 92 B per box -> pure HBM-bandwidth problem (~184 MB,
// ~8us at 23.3 TB/s). No matmul exists in this op, so WMMA is not applicable;
// the CDNA5-specific win is the async global->LDS DMA path (ASYNCcnt) to turn
// the 20-byte-stride box records into pure b128 global traffic, plus
// non-temporal b128 stores so the 96 MB of streamed output doesn't pollute L2.
//
// Steady-state blocks use fully-unrolled async copies with immediate offsets
// (ISA: INST_OFFSET is added to BOTH the LDS and global address, so one
// (lds,global) register pair + `offset:` immediates covers a whole region).
// ---------------------------------------------------------------------------

typedef float v4f __attribute__((ext_vector_type(4)));

#define THREADS 256   // 8 wave32 per block
#define BPB     1024  // boxes per block: rois/gt tiles = 20480 B, labels = 4096 B

// Generic (tail-block) async copy: bulk b128 + b32 remainder, never OOB.
__device__ __forceinline__ void async_copy_to_lds(uint32_t lds_base,
                                                  const void* gsrc,
                                                  int bytes, int tid) {
  const char* g = (const char*)gsrc;
  const int full = bytes & ~15;
  for (int off = tid * 16; off < full; off += THREADS * 16) {
    uint64_t ga = (uint64_t)(uintptr_t)(g + off);
    uint32_t la = lds_base + (uint32_t)off;
    asm volatile("global_load_async_to_lds_b128 %0, %1, off"
                 :: "v"(la), "v"(ga) : "memory");
  }
  const int rem = bytes - full;           // 0..12 bytes, multiple of 4
  if (tid * 4 < rem) {
    uint64_t ga = (uint64_t)(uintptr_t)(g + full + tid * 4);
    uint32_t la = lds_base + (uint32_t)(full + tid * 4);
    asm volatile("global_load_async_to_lds_b32 %0, %1, off"
                 :: "v"(la), "v"(ga) : "memory");
  }
}

// Full-tile copy of a 20480-byte region: 5 unrolled b128 DMAs per lane,
// stride = THREADS*16 = 4096 bytes, folded into the instruction offset.
__device__ __forceinline__ void async_copy_20480(uint32_t lds_base,
                                                 const void* gsrc, int tid) {
  uint64_t ga = (uint64_t)(uintptr_t)((const char*)gsrc + tid * 16);
  uint32_t la = lds_base + (uint32_t)(tid * 16);
  asm volatile("global_load_async_to_lds_b128 %0, %1, off"
               :: "v"(la), "v"(ga) : "memory");
  asm volatile("global_load_async_to_lds_b128 %0, %1, off offset:4096"
               :: "v"(la), "v"(ga) : "memory");
  asm volatile("global_load_async_to_lds_b128 %0, %1, off offset:8192"
               :: "v"(la), "v"(ga) : "memory");
  asm volatile("global_load_async_to_lds_b128 %0, %1, off offset:12288"
               :: "v"(la), "v"(ga) : "memory");
  asm volatile("global_load_async_to_lds_b128 %0, %1, off offset:16384"
               :: "v"(la), "v"(ga) : "memory");
}

// Full-tile copy of a 4096-byte region: 1 b128 DMA per lane.
__device__ __forceinline__ void async_copy_4096(uint32_t lds_base,
                                                const void* gsrc, int tid) {
  uint64_t ga = (uint64_t)(uintptr_t)((const char*)gsrc + tid * 16);
  uint32_t la = lds_base + (uint32_t)(tid * 16);
  asm volatile("global_load_async_to_lds_b128 %0, %1, off"
               :: "v"(la), "v"(ga) : "memory");
}

__global__ __launch_bounds__(THREADS)
void rcnn_target_gfx1250(const float* __restrict__ gt_rois,   // [N,5] (x1,y1,x2,y2,cls)
                         const float* __restrict__ rois,      // [N,5] (b,x1,y1,x2,y2)
                         const int*   __restrict__ labels,    // [N]
                         const float* __restrict__ means,     // [4]
                         const float* __restrict__ stds,      // [4]
                         const float* __restrict__ inside_w,  // [4]
                         float* __restrict__ out,             // [3][N,4] concat
                         int n) {
  __shared__ float s_gt [BPB * 5];
  __shared__ float s_roi[BPB * 5];
  __shared__ int   s_lab[BPB];
  __shared__ float s_par[12];   // means[4], 1/std[4], inside_w[4]

  const int tid  = threadIdx.x;
  const int base = (int)blockIdx.x * BPB;
  if (base >= n) return;
  const int nb = min(BPB, n - base);

  // LDS byte addresses: generic LDS pointers carry the LDS offset in the low
  // 32 bits (aperture encoding), which is what the async DMA vdst expects.
  const uint32_t lds_gt  = (uint32_t)(uintptr_t)s_gt;
  const uint32_t lds_roi = (uint32_t)(uintptr_t)s_roi;
  const uint32_t lds_lab = (uint32_t)(uintptr_t)s_lab;
  const float* g_gt  = gt_rois + (size_t)base * 5;
  const float* g_roi = rois    + (size_t)base * 5;
  const int*   g_lab = labels  + base;

  if (nb == BPB) {
    // Steady state: fully unrolled, compile-time offsets, 11 DMAs per lane.
    async_copy_20480(lds_gt,  g_gt,  tid);
    async_copy_20480(lds_roi, g_roi, tid);
    async_copy_4096 (lds_lab, g_lab, tid);
  } else {
    async_copy_to_lds(lds_gt,  g_gt,  nb * 5 * 4, tid);
    async_copy_to_lds(lds_roi, g_roi, nb * 5 * 4, tid);
    async_copy_to_lds(lds_lab, g_lab, nb * 4,     tid);
  }

  if (tid < 4) {
    s_par[tid]     = means[tid];
    s_par[4 + tid] = 1.0f / stds[tid];
    s_par[8 + tid] = inside_w[tid];
  }

  // Drain this wave's ASYNCcnt, then publish LDS to the whole workgroup.
#if __has_builtin(__builtin_amdgcn_s_wait_asynccnt)
  __builtin_amdgcn_s_wait_asynccnt(0);
#else
  asm volatile("s_wait_asynccnt 0" ::: "memory");
#endif
  __syncthreads();

  const float m0 = s_par[0], m1 = s_par[1], m2 = s_par[2],  m3 = s_par[3];
  const float i0 = s_par[4], i1 = s_par[5], i2 = s_par[6],  i3 = s_par[7];
  const float w0 = s_par[8], w1 = s_par[9], w2 = s_par[10], w3 = s_par[11];

  float* __restrict__ out_t = out;                      // bbox_targets
  float* __restrict__ out_i = out + (size_t)n * 4;      // bbox_inside_weights
  float* __restrict__ out_o = out + (size_t)n * 8;      // bbox_outside_weights

#pragma unroll
  for (int j = 0; j < 4; ++j) {
    const int l = tid * 4 + j;                          // local box index
    if (l >= nb) break;
    const int box = base + l;

    const float gx1 = s_gt[l * 5 + 0], gy1 = s_gt[l * 5 + 1];
    const float gx2 = s_gt[l * 5 + 2], gy2 = s_gt[l * 5 + 3];
    const float rx1 = s_roi[l * 5 + 1], ry1 = s_roi[l * 5 + 2];
    const float rx2 = s_roi[l * 5 + 3], ry2 = s_roi[l * 5 + 4];
    const bool  pos = s_lab[l] > 0;

    const float ex_w  = rx2 - rx1 + 1.0f;
    const float ex_h  = ry2 - ry1 + 1.0f;
    const float ex_cx = rx1 + 0.5f * ex_w;
    const float ex_cy = ry1 + 0.5f * ex_h;
    const float gt_w  = gx2 - gx1 + 1.0f;
    const float gt_h  = gy2 - gy1 + 1.0f;
    const float gt_cx = gx1 + 0.5f * gt_w;
    const float gt_cy = gy1 + 0.5f * gt_h;

    const float inv_w = 1.0f / ex_w;
    const float inv_h = 1.0f / ex_h;
    const float dx = (gt_cx - ex_cx) * inv_w;
    const float dy = (gt_cy - ex_cy) * inv_h;
    const float dw = __logf(gt_w * inv_w);
    const float dh = __logf(gt_h * inv_h);

    v4f tgt;
    tgt.x = pos ? (dx - m0) * i0 : 0.0f;
    tgt.y = pos ? (dy - m1) * i1 : 0.0f;
    tgt.z = pos ? (dw - m2) * i2 : 0.0f;
    tgt.w = pos ? (dh - m3) * i3 : 0.0f;

    v4f inw;
    inw.x = pos ? w0 : 0.0f;
    inw.y = pos ? w1 : 0.0f;
    inw.z = pos ? w2 : 0.0f;
    inw.w = pos ? w3 : 0.0f;

    v4f onw;
    onw.x = inw.x > 0.0f ? 1.0f : 0.0f;
    onw.y = inw.y > 0.0f ? 1.0f : 0.0f;
    onw.z = inw.z > 0.0f ? 1.0f : 0.0f;
    onw.w = inw.w > 0.0f ? 1.0f : 0.0f;

    // Streamed once, never re-read: non-temporal b128 stores.
    __builtin_nontemporal_store(tgt, (v4f*)(out_t + (size_t)box * 4));
    __builtin_nontemporal_store(inw, (v4f*)(out_i + (size_t)box * 4));
    __builtin_nontemporal_store(onw, (v4f*)(out_o + (size_t)box * 4));
  }
}

extern "C" void kernel_launch(void* const* d_in, const int* in_sizes, int n_in,
                              void* d_out, int out_size, void* d_ws, size_t ws_size,
                              hipStream_t stream) {
  (void)n_in; (void)out_size; (void)d_ws; (void)ws_size;
  const float* gt_rois  = (const float*)d_in[0];  // [1,N,5]
  const float* rois     = (const float*)d_in[1];  // [1,N,5]
  const int*   labels   = (const int*)d_in[2];    // [N]
  const float* means    = (const float*)d_in[3];  // [4]
  const float* stds     = (const float*)d_in[4];  // [4]
  const float* inside_w = (const float*)d_in[5];  // [4]
  float* out = (float*)d_out;                     // 3 x [1,N,4] concatenated

  const int n = in_sizes[2];                      // N from labels
  const int blocks = (n + BPB - 1) / BPB;
  rcnn_target_gfx1250<<<blocks, THREADS, 0, stream>>>(
      gt_rois, rois, labels, means, stds, inside_w, out, n);
}